// CustomRouterRouter_43739946942520
// MI455X (gfx1250) — compile-verified
//
#include <hip/hip_runtime.h>

// ---------------------------------------------------------------------------
// CustomRouter for MI455X (gfx1250, wave32).
//
// out[l,e] = rr_e(l) + rr2(l) * linn[l,e],  linn = rownorm(x @ W^T + b)
//
// Roofline: streaming x (128 MiB) dominates -> ~5.6 us floor @ 23.3 TB/s.
// Compute (134 MFLOP) is negligible -> exact-precision V_WMMA_F32_16X16X4_F32.
//
// Concurrency: need ~9 MB in flight (23.3 TB/s x ~400 ns).
//   * split-K: each 16-row tile's H=4096 reduction divided across 4 waves
//     (chunks of 1024) -> 2048 waves, partials combined through LDS.
//   * The backend scheduler only keeps ~2 loads in flight when loads feed
//     WMMAs directly (the acc chain hides no memory latency), so we
//     software-pipeline in source: explicit register double-buffering of
//     the x stream, issuing all 16 b64 loads of the NEXT 64-element group
//     before consuming the current one -> >=16 outstanding loads (4 KB)
//     per wave, ~8 MB chip-wide, plus global_prefetch_b8 lookahead.
//
// B columns >= 2 are junk (clamped reads of W row 0): WMMA outputs are
// independent dot products, so unused D columns never contaminate the two
// real ones -- no masking VALU work in the hot loop.
// ---------------------------------------------------------------------------

typedef float v2f __attribute__((ext_vector_type(2)));
typedef float v8f __attribute__((ext_vector_type(8)));

#define L_DIM   8192
#define H_DIM   4096
#define B_DIM   2
#define N_DIM   2
#define E_DIM   2
#define TILE_M  16
#define SPLITK  4
#define H_CHUNK (H_DIM / SPLITK)               // 1024 per split wave
#define TILES_PER_BLOCK 2
#define WAVES_PER_BLOCK (TILES_PER_BLOCK * SPLITK)   // 8
#define THREADS (WAVES_PER_BLOCK * 32)               // 256
#define ROWS_PER_SEG  (L_DIM / B_DIM)          // 4096 rows share one hs
#define TILES_PER_SEG (ROWS_PER_SEG / TILE_M)  // 256 (4096 % 16 == 0)
#define NBLOCKS (L_DIM / (TILE_M * TILES_PER_BLOCK)) // 256

#define G  16          // WMMAs per pipeline group
#define GH (G * 4)     // 64 H-elements per group; H_CHUNK/GH = 16 groups

__global__ __launch_bounds__(THREADS, 1)
void router_kernel(const float* __restrict__ x,
                   const float* __restrict__ weights,
                   const float* __restrict__ W,
                   const float* __restrict__ bvec,
                   float* __restrict__ out)
{
    // ---- LDS: W (2 x 4096 f32 = 32 KiB) + split-K reduction scratch ------
    __shared__ float Wlds[E_DIM * H_DIM];
    __shared__ float red[TILES_PER_BLOCK][SPLITK - 1][TILE_M * E_DIM]; // 768 B

    {
        const float4* src = (const float4*)W;
        float4*       dst = (float4*)Wlds;
        const int n4 = (E_DIM * H_DIM) / 4;  // 2048, uniform trip count
        for (int i = threadIdx.x; i < n4; i += THREADS) dst[i] = src[i];
    }
    __syncthreads();

    const int lane    = threadIdx.x & 31;
    const int wave    = threadIdx.x >> 5;
    const int split   = wave & (SPLITK - 1);          // K-chunk id, 0..3
    const int tileBlk = wave >> 2;                    // tile within block
    const int tile    = blockIdx.x * TILES_PER_BLOCK + tileBlk;  // 0..511
    const int rowBase = tile * TILE_M;
    const int hBase   = split * H_CHUNK;

    // ISA WMMA f32 16x16x4 operand layout (wave32):
    //   A (16x4): lane t holds row M = t%16; VGPR0/1 = K = s, s+1 with
    //             s = (t<16) ? 0 : 2  -> contiguous float2 from x.
    //   B (4x16): lane t holds col N = t%16; VGPR0/1 = K = s, s+1.
    //             Cols 0,1 = rows of W; cols >= 2 read clamped (junk is OK,
    //             those D columns are never consumed).
    const int n    = lane & 15;
    const int s    = (lane < 16) ? 0 : 2;
    const int eIdx = (n < 2) ? n : 0;                 // clamp: safe address

    const float* aptr = x + (size_t)(rowBase + n) * H_DIM + hBase + s;
    const float* bptr = &Wlds[eIdx * H_DIM + hBase + s];

    v8f acc = {};  // C/D 16x16 f32 accumulator (8 VGPRs)

    // ---- explicit register double-buffered pipeline over the x stream ----
    auto loadA = [&](v2f (&buf)[G], int h) {
        #pragma unroll
        for (int g = 0; g < G; ++g)
            buf[g] = *(const v2f*)(aptr + h + 4 * g);   // global_load_b64
    };
    auto consume = [&](v2f (&buf)[G], int h) {
        __builtin_prefetch(aptr + h + 512, 0, 1);       // ~2 KiB lookahead
        #pragma unroll
        for (int g = 0; g < G; ++g) {
            v2f bb = *(const v2f*)(bptr + h + 4 * g);   // ds_load_b64
            // D = A(16x4) * B(4x16) + C, f32 (exact vs reference)
            acc = __builtin_amdgcn_wmma_f32_16x16x4_f32(
                /*neg_a=*/false, buf[g], /*neg_b=*/false, bb,
                /*c_mod=*/(short)0, acc,
                /*reuse_a=*/false, /*reuse_b=*/false);
        }
    };

    v2f buf0[G], buf1[G];
    loadA(buf0, 0);
    int h = 0;
    for (; h + 2 * GH < H_CHUNK; h += 2 * GH) {  // period-2: no dynamic index
        loadA(buf1, h + GH);        // issue next group's 16 loads first
        consume(buf0, h);           // then run current group's WMMA chain
        loadA(buf0, h + 2 * GH);
        consume(buf1, h + GH);
    }
    loadA(buf1, h + GH);            // h == H_CHUNK - 2*GH here
    consume(buf0, h);
    consume(buf1, h + GH);

    // D layout: VGPR r -> M = r (lanes 0-15) / r+8 (lanes 16-31), N = lane%16
    const int mBase = (lane < 16) ? 0 : 8;

    // ---- split-K reduction through LDS ------------------------------------
    if (split != 0) {
        #pragma unroll
        for (int r = 0; r < 8; ++r) {
            if (n < 2)
                red[tileBlk][split - 1][(mBase + r) * E_DIM + n] = acc[r];
        }
    }
    __syncthreads();
    if (split != 0) return;

    #pragma unroll
    for (int r = 0; r < 8; ++r) {
        if (n < 2) {
            const int idx = (mBase + r) * E_DIM + n;
            acc[r] += red[tileBlk][0][idx] + red[tileBlk][1][idx]
                    + red[tileBlk][2][idx];
        }
    }

    // ---- epilogue: routing weights (wave-uniform: a tile never straddles a
    //      4096-row segment), row-normalize, combine, store ----
    const int   seg = tile / TILES_PER_SEG;
    const float hs  = weights[seg * N_DIM + 0];
    const bool  hi  = (hs >= 0.5f);
    const float rr0 = hi ? (2.0f * hs - 1.0f) : 0.0f;
    const float rr1 = hi ? 0.0f : (1.0f - 2.0f * hs);
    const float rr2 = hi ? (2.0f - 2.0f * hs) : (2.0f * hs);
    const float b0  = bvec[0];
    const float b1  = bvec[1];

    const float be  = (n == 0) ? b0 : b1;   // bias for own column
    const float bo  = (n == 0) ? b1 : b0;   // bias for sibling column
    const float rre = (n == 0) ? rr0 : rr1;

    #pragma unroll
    for (int r = 0; r < 8; ++r) {
        float v     = acc[r];
        float other = __shfl_xor(v, 1, 32);   // pair e=0 <-> e=1 (all lanes)
        if (n < 2) {
            float lin  = v + be;
            float linO = other + bo;
            float o    = rre + rr2 * (lin / (lin + linO));
            out[(size_t)(rowBase + mBase + r) * E_DIM + n] = o;
        }
    }
}

extern "C" void kernel_launch(void* const* d_in, const int* in_sizes, int n_in,
                              void* d_out, int out_size, void* d_ws, size_t ws_size,
                              hipStream_t stream) {
    (void)in_sizes; (void)n_in; (void)out_size; (void)d_ws; (void)ws_size;
    const float* x       = (const float*)d_in[0];  // (8192, 4096)
    const float* weights = (const float*)d_in[1];  // (2, 2)
    const float* W       = (const float*)d_in[2];  // (2, 4096)
    const float* b       = (const float*)d_in[3];  // (2,)
    float*       out     = (float*)d_out;          // (8192, 2)

    hipLaunchKernelGGL(router_kernel, dim3(NBLOCKS), dim3(THREADS), 0, stream,
                       x, weights, W, b, out);
}